// Repeat_Recommendation_Decoder_28716151341089
// MI455X (gfx1250) — compile-verified
//
#include <hip/hip_runtime.h>
#include <math.h>

typedef __attribute__((ext_vector_type(2))) float v2f;
typedef __attribute__((ext_vector_type(8))) float v8f;

#define HIDDEN 64
#define SEQ    50
#define NITEM  40000

__global__ __launch_bounds__(256)
void repeat_rec_decoder(const float* __restrict__ all_memory,
                        const float* __restrict__ last_memory,
                        const int*   __restrict__ seq_item,
                        const float* __restrict__ Wr,
                        const float* __restrict__ Ur,
                        const float* __restrict__ Vr_w,
                        const float* __restrict__ Vr_b,
                        float* __restrict__ out)
{
    __shared__ float s_scores[64];
    __shared__ float s_probs[64];

    const int b    = blockIdx.x;
    const int tid  = threadIdx.x;
    const int wave = tid >> 5;
    const int lane = tid & 31;

    float* out_row = out + (size_t)b * NITEM;

    if (wave != 0) {
        // ---- waves 1..7: zero the 160 KB output row with b128 stores ----
        float4* row4 = (float4*)out_row;
        const float4 z = make_float4(0.f, 0.f, 0.f, 0.f);
        for (int i = tid - 32; i < NITEM / 4; i += 224) row4[i] = z;
    } else {
        // ---- wave 0: attention math via V_WMMA_F32_16X16X4_F32 ----
        const float* am = all_memory  + (size_t)b * SEQ * HIDDEN;
        const float* lm = last_memory + (size_t)b * HIDDEN;

        const int n  = lane & 15;      // N (and M) index within 16-wide tile
        const int hi = lane >> 4;      // half-wave select
        const int kb = hi * 2;         // K sub-offset within group of 4

        // l[k] = last_memory . Wr[k,:]  and Vr_w[k] for this lane's columns
        float lv[4], vw[4];
#pragma unroll
        for (int nt = 0; nt < 4; ++nt) {
            const int k = nt * 16 + n;
            float acc = 0.f;
#pragma unroll
            for (int h = 0; h < HIDDEN; ++h)
                acc = fmaf(lm[h], Wr[k * HIDDEN + h], acc);
            lv[nt] = acc;
            vw[nt] = Vr_w[k];
        }

        float sc[4][8];
#pragma unroll
        for (int mt = 0; mt < 4; ++mt)
#pragma unroll
            for (int r = 0; r < 8; ++r) sc[mt][r] = 0.f;

#pragma unroll
        for (int nt = 0; nt < 4; ++nt) {
            // B fragment: B[K][N] = Ur[nt*16+N, kk+K]
            // layout: v0 = K={0|2}, v1 = K={1|3}   (lanes 0-15 | 16-31)
            float breg[32];
#pragma unroll
            for (int kk = 0; kk < 16; ++kk) {
                breg[2 * kk + 0] = Ur[(nt * 16 + n) * HIDDEN + 4 * kk + kb + 0];
                breg[2 * kk + 1] = Ur[(nt * 16 + n) * HIDDEN + 4 * kk + kb + 1];
            }
#pragma unroll
            for (int mt = 0; mt < 4; ++mt) {
                const int  row = mt * 16 + n;
                const bool rv  = row < SEQ;
                // A fragment: A[M][K] = all_memory[b, row, kk+K]
                float areg[32];
#pragma unroll
                for (int kk = 0; kk < 16; ++kk) {
                    areg[2 * kk + 0] = rv ? am[row * HIDDEN + 4 * kk + kb + 0] : 0.f;
                    areg[2 * kk + 1] = rv ? am[row * HIDDEN + 4 * kk + kb + 1] : 0.f;
                }
                v8f acc = {0.f, 0.f, 0.f, 0.f, 0.f, 0.f, 0.f, 0.f};
#pragma unroll
                for (int kk = 0; kk < 16; ++kk) {
                    v2f av; av[0] = areg[2 * kk]; av[1] = areg[2 * kk + 1];
                    v2f bv; bv[0] = breg[2 * kk]; bv[1] = breg[2 * kk + 1];
                    acc = __builtin_amdgcn_wmma_f32_16x16x4_f32(
                        /*neg_a=*/false, av, /*neg_b=*/false, bv,
                        /*c_mod=*/(short)0, acc,
                        /*reuse_a=*/false, /*reuse_b=*/false);
                }
                // fold tile into per-lane partial scores: tanh(a+l)*vw
#pragma unroll
                for (int r = 0; r < 8; ++r)
                    sc[mt][r] += tanhf(acc[r] + lv[nt]) * vw[nt];
            }
        }

        const float vrb = Vr_b[0];
        // reduce each sc[mt][r] across its 16-lane group (k-dimension)
#pragma unroll
        for (int mt = 0; mt < 4; ++mt) {
#pragma unroll
            for (int r = 0; r < 8; ++r) {
                float v = sc[mt][r];
                v += __shfl_xor(v, 1, 32);
                v += __shfl_xor(v, 2, 32);
                v += __shfl_xor(v, 4, 32);
                v += __shfl_xor(v, 8, 32);
                if (n == 0) s_scores[mt * 16 + hi * 8 + r] = v + vrb;
            }
        }
    }

    __syncthreads();

    if (wave == 0) {
        // ---- softmax over SEQ=50 scores (lane covers s and s+32) ----
        float x0 = (lane < SEQ)      ? s_scores[lane]      : -INFINITY;
        float x1 = (lane + 32 < SEQ) ? s_scores[lane + 32] : -INFINITY;
        float m = fmaxf(x0, x1);
#pragma unroll
        for (int o = 1; o < 32; o <<= 1) m = fmaxf(m, __shfl_xor(m, o, 32));
        float e0 = (lane < SEQ)      ? __expf(x0 - m) : 0.f;
        float e1 = (lane + 32 < SEQ) ? __expf(x1 - m) : 0.f;
        float s = e0 + e1;
#pragma unroll
        for (int o = 1; o < 32; o <<= 1) s += __shfl_xor(s, o, 32);
        float inv = __frcp_rn(s);
        if (lane < SEQ)      s_probs[lane]      = e0 * inv;
        if (lane + 32 < SEQ) s_probs[lane + 32] = e1 * inv;
    } else {
        __threadfence();   // zero-fill stores visible before scatter RMWs
    }

    __syncthreads();

    // ---- sparse scatter: 50 f32 atomics into the block-owned row ----
    if (tid < SEQ) {
        const int item = seq_item[b * SEQ + tid];
        atomicAdd(out_row + item, s_probs[tid]);
    }
}

extern "C" void kernel_launch(void* const* d_in, const int* in_sizes, int n_in,
                              void* d_out, int out_size, void* d_ws, size_t ws_size,
                              hipStream_t stream) {
    const float* all_memory  = (const float*)d_in[0];
    const float* last_memory = (const float*)d_in[1];
    const int*   seq_item    = (const int*)  d_in[2];
    const float* Wr          = (const float*)d_in[3];
    const float* Ur          = (const float*)d_in[4];
    const float* Vr_w        = (const float*)d_in[5];
    const float* Vr_b        = (const float*)d_in[6];
    float*       out         = (float*)d_out;

    const int B = in_sizes[1] / HIDDEN;   // 1024
    repeat_rec_decoder<<<B, 256, 0, stream>>>(all_memory, last_memory, seq_item,
                                              Wr, Ur, Vr_w, Vr_b, out);
}